// GraphEncoder_17205638988259
// MI455X (gfx1250) — compile-verified
//
#include <hip/hip_runtime.h>
#include <stdint.h>

typedef __attribute__((ext_vector_type(2))) float v2f;
typedef __attribute__((ext_vector_type(8))) float v8f;

// ---------------------------------------------------------------- utilities
__global__ void zero_f32(float* __restrict__ p, long long n) {
  long long i = (long long)blockIdx.x * blockDim.x + threadIdx.x;
  if (i < n) p[i] = 0.0f;
}

// deg[dst] += 1 over all E edges (self-loop +1 added in dinv_kernel)
__global__ void degree_kernel(const long long* __restrict__ dst,
                              float* __restrict__ degf, long long E) {
  long long e = (long long)blockIdx.x * blockDim.x + threadIdx.x;
  if (e < E) atomicAdd(&degf[dst[e]], 1.0f);
}

__global__ void dinv_kernel(const float* __restrict__ degf,
                            float* __restrict__ dinv, long long N) {
  long long i = (long long)blockIdx.x * blockDim.x + threadIdx.x;
  if (i < N) dinv[i] = rsqrtf(degf[i] + 1.0f);  // +1 self-loop; always > 0
}

// ------------------------------------------------- layer 1 linear: h1 = x @ W1
// x:[N,4]  W1:[4,16]  h1:[N,16].  One wave = one 16-row tile, one fp32 WMMA
// with K = 4 = Fin (V_WMMA_F32_16X16X4_F32).
__global__ void linear1_wmma(const float* __restrict__ x,
                             const float* __restrict__ W1,
                             float* __restrict__ h1, long long N) {
  int lane = threadIdx.x & 31;
  long long tile = ((long long)blockIdx.x * blockDim.x + threadIdx.x) >> 5;
  long long base = tile * 16;
  if (base >= N) return;  // wave-uniform exit

  int m  = lane & 15;            // A row / B column held by this lane
  int kb = (lane >> 4) << 1;     // K pair: lanes 0-15 -> K=0,1 ; 16-31 -> K=2,3

  long long row = base + m; if (row >= N) row = N - 1;  // clamp (branch-free)
  v2f a; a.x = x[row * 4 + kb];      a.y = x[row * 4 + kb + 1];
  v2f b; b.x = W1[kb * 16 + m];      b.y = W1[(kb + 1) * 16 + m];

  v8f c = {};
  c = __builtin_amdgcn_wmma_f32_16x16x4_f32(false, a, false, b,
                                            (short)0, c, false, false);

  // C/D layout: VGPR r -> M = r + 8*(lane>=16), N = lane&15
  int mb = (lane >> 4) << 3;
  float* p = h1 + (base + mb) * 16 + m;      // row stride = 16 floats = 64 B
  if (base + 16 <= N) {                      // wave-uniform: full tile, no guards
    #pragma unroll
    for (int r = 0; r < 8; ++r) p[(long long)r * 16] = c[r];
  } else {                                   // tail tile: per-row guard
    #pragma unroll
    for (int r = 0; r < 8; ++r)
      if (base + mb + r < N) p[(long long)r * 16] = c[r];
  }
}

// ---------------------------------------- layer 1 edge scatter (atomic add)
// one lane per (edge, feature): agg[dst,f] += h1[src,f] * dinv[src]*dinv[dst]
__global__ void scatter1(const long long* __restrict__ srcs,
                         const long long* __restrict__ dsts,
                         const float* __restrict__ dinv,
                         const float* __restrict__ h1,
                         float* __restrict__ agg, long long E) {
  long long t = (long long)blockIdx.x * blockDim.x + threadIdx.x;
  if (t >= (E << 4)) return;
  long long e = t >> 4;
  int f = (int)(t & 15);
  long long s = srcs[e], d = dsts[e];
  float w = dinv[s] * dinv[d];
  atomicAdd(&agg[d * 16 + f], h1[s * 16 + f] * w);
}

// self-loop message + bias + ReLU, in place: agg <- relu(agg + h1*dinv^2 + b1)
__global__ void finalize1(const float* __restrict__ h1,
                          const float* __restrict__ dinv,
                          const float* __restrict__ b1,
                          float* __restrict__ agg, long long N) {
  long long t = (long long)blockIdx.x * blockDim.x + threadIdx.x;
  if (t >= N * 16) return;
  long long i = t >> 4;
  int f = (int)(t & 15);
  float di = dinv[i];
  float v = agg[t] + h1[t] * di * di + b1[f];
  agg[t] = v > 0.0f ? v : 0.0f;
}

// ------------------------------------------- layer 2 linear: h2 = hrelu @ W2
// hrelu:[N,16]  W2:[16,8] (zero-padded to 16 cols in-lane)  h2:[N,8]
// K = 16 -> four chained V_WMMA_F32_16X16X4_F32.
__global__ void linear2_wmma(const float* __restrict__ hr,
                             const float* __restrict__ W2,
                             float* __restrict__ h2, long long N) {
  int lane = threadIdx.x & 31;
  long long tile = ((long long)blockIdx.x * blockDim.x + threadIdx.x) >> 5;
  long long base = tile * 16;
  if (base >= N) return;

  int m  = lane & 15;
  int kb = (lane >> 4) << 1;
  long long row = base + m; if (row >= N) row = N - 1;
  int cn = m < 8 ? m : 7;            // clamped W2 column (branch-free pad)

  v8f c = {};
  #pragma unroll
  for (int ks = 0; ks < 4; ++ks) {
    int k0 = ks * 4 + kb;
    v2f a; a.x = hr[row * 16 + k0];  a.y = hr[row * 16 + k0 + 1];
    float bx = W2[k0 * 8 + cn];
    float by = W2[(k0 + 1) * 8 + cn];
    if (m >= 8) { bx = 0.0f; by = 0.0f; }   // columns 8..15 of padded W2
    v2f b; b.x = bx; b.y = by;
    c = __builtin_amdgcn_wmma_f32_16x16x4_f32(false, a, false, b,
                                              (short)0, c, false, false);
  }

  int mb = (lane >> 4) << 3;
  float* p = h2 + (base + mb) * 8 + m;       // row stride = 8 floats = 32 B
  if (base + 16 <= N) {                      // wave-uniform full-tile fast path
    if (m < 8) {                             // only first 8 output columns real
      #pragma unroll
      for (int r = 0; r < 8; ++r) p[(long long)r * 8] = c[r];
    }
  } else {
    if (m < 8) {
      #pragma unroll
      for (int r = 0; r < 8; ++r)
        if (base + mb + r < N) p[(long long)r * 8] = c[r];
    }
  }
}

// ---------------------------------------- layer 2 edge scatter (atomic add)
__global__ void scatter2(const long long* __restrict__ srcs,
                         const long long* __restrict__ dsts,
                         const float* __restrict__ dinv,
                         const float* __restrict__ h2,
                         float* __restrict__ out, long long E) {
  long long t = (long long)blockIdx.x * blockDim.x + threadIdx.x;
  if (t >= (E << 3)) return;
  long long e = t >> 3;
  int f = (int)(t & 7);
  long long s = srcs[e], d = dsts[e];
  float w = dinv[s] * dinv[d];
  atomicAdd(&out[d * 8 + f], h2[s * 8 + f] * w);
}

// out += h2*dinv^2 + b2
__global__ void finalize2(const float* __restrict__ h2,
                          const float* __restrict__ dinv,
                          const float* __restrict__ b2,
                          float* __restrict__ out, long long N) {
  long long t = (long long)blockIdx.x * blockDim.x + threadIdx.x;
  if (t >= N * 8) return;
  long long i = t >> 3;
  int f = (int)(t & 7);
  float di = dinv[i];
  out[t] = out[t] + h2[t] * di * di + b2[f];
}

// ---------------------------------------------------------------- launcher
static inline unsigned blocks_for(long long n, int bs) {
  return (unsigned)((n + bs - 1) / bs);
}

extern "C" void kernel_launch(void* const* d_in, const int* in_sizes, int n_in,
                              void* d_out, int out_size, void* d_ws, size_t ws_size,
                              hipStream_t stream) {
  const float*     x   = (const float*)d_in[0];
  const long long* ei  = (const long long*)d_in[1];   // [2,E] row-major, int64
  const float*     W1  = (const float*)d_in[2];
  const float*     b1  = (const float*)d_in[3];
  const float*     W2  = (const float*)d_in[4];
  const float*     b2  = (const float*)d_in[5];
  float*           out = (float*)d_out;

  long long N = in_sizes[0] / 4;   // Fin = 4
  long long E = in_sizes[1] / 2;
  const long long* srcs = ei;
  const long long* dsts = ei + E;

  // workspace layout (floats): degf[N] | dinv[N] | h1[16N] | agg[16N] | h2[8N]
  float* ws   = (float*)d_ws;
  float* degf = ws;
  float* dinv = ws + N;
  float* h1   = ws + 2 * N;
  float* agg  = ws + 18 * N;
  float* h2   = ws + 34 * N;

  const int BS = 256;

  // zero accumulation buffers (harness poisons once; we replay under graphs,
  // so every call must re-initialize its own accumulators)
  zero_f32<<<blocks_for(N, BS),      BS, 0, stream>>>(degf, N);
  zero_f32<<<blocks_for(N * 16, BS), BS, 0, stream>>>(agg,  N * 16);
  zero_f32<<<blocks_for(N * 8, BS),  BS, 0, stream>>>(out,  N * 8);

  // degrees + normalization
  degree_kernel<<<blocks_for(E, BS), BS, 0, stream>>>(dsts, degf, E);
  dinv_kernel<<<blocks_for(N, BS),   BS, 0, stream>>>(degf, dinv, N);

  // layer 1
  long long tiles = (N + 15) / 16;          // one wave per 16-node tile
  linear1_wmma<<<blocks_for(tiles * 32, BS), BS, 0, stream>>>(x, W1, h1, N);
  scatter1<<<blocks_for(E * 16, BS), BS, 0, stream>>>(srcs, dsts, dinv, h1, agg, E);
  finalize1<<<blocks_for(N * 16, BS), BS, 0, stream>>>(h1, dinv, b1, agg, N);

  // layer 2
  linear2_wmma<<<blocks_for(tiles * 32, BS), BS, 0, stream>>>(agg, W2, h2, N);
  scatter2<<<blocks_for(E * 8, BS), BS, 0, stream>>>(srcs, dsts, dinv, h2, out, E);
  finalize2<<<blocks_for(N * 8, BS), BS, 0, stream>>>(h2, dinv, b2, out, N);
}